// ImageInterpolator_18691697672363
// MI455X (gfx1250) — compile-verified
//
#include <hip/hip_runtime.h>

// Bilinear crop-resample: image (64,512,512,8) f32, section (64,3) f32
// -> out (64,64,64,8) f32.  Memory-bound gather; optimized for b128
// vector loads, scalar section loads, wave32.

#define IMG_H 512
#define IMG_W 512
#define IMG_C 8
#define GRID0 64
#define GRID1 64

__device__ __forceinline__ float4 lerp4(float4 a, float4 b, float w, float omw) {
    float4 r;
    r.x = fmaf(b.x, w, a.x * omw);
    r.y = fmaf(b.y, w, a.y * omw);
    r.z = fmaf(b.z, w, a.z * omw);
    r.w = fmaf(b.w, w, a.w * omw);
    return r;
}

__global__ __launch_bounds__(256, 4)
void bilinear_crop_kernel(const float* __restrict__ image,
                          const float* __restrict__ section,
                          float* __restrict__ out) {
    const int tid  = threadIdx.x;
    const int x    = tid & 63;        // grid dim 1 (fast axis, coalesced stores)
    const int yl   = tid >> 6;        // 0..3
    const int tile = blockIdx.x;      // 0..1023
    const int b    = tile >> 4;       // batch: block-uniform -> scalar loads
    const int y    = ((tile & 15) << 2) + yl;

    // section[b] = (start0, start1, len); stop = start + len for both dims
    const float s0  = section[b * 3 + 0];
    const float s1  = section[b * 3 + 1];
    const float len = section[b * 3 + 2];

    const float inv63 = 1.0f / 63.0f;
    const float q0 = fmaf((float)y * inv63, len, s0);
    const float q1 = fmaf((float)x * inv63, len, s1);

    float i0 = q0 * (float)(IMG_H - 1);
    float i1 = q1 * (float)(IMG_W - 1);
    i0 = fminf(fmaxf(i0, 0.0f), (float)(IMG_H - 1));
    i1 = fminf(fmaxf(i1, 0.0f), (float)(IMG_W - 1));

    const float f0 = floorf(i0);
    const float f1 = floorf(i1);
    const float w0 = i0 - f0;
    const float w1 = i1 - f1;
    const float omw0 = 1.0f - w0;
    const float omw1 = 1.0f - w1;

    const int lo0 = (int)f0;
    const int lo1 = (int)f1;
    const int hi0 = min(lo0 + 1, IMG_H - 1);
    const int hi1 = min(lo1 + 1, IMG_W - 1);

    const size_t imgBase = (size_t)b * IMG_H * IMG_W * IMG_C;
    const float* rowLo = image + imgBase + (size_t)lo0 * IMG_W * IMG_C;
    const float* rowHi = image + imgBase + (size_t)hi0 * IMG_W * IMG_C;

    const float4* p00 = (const float4*)(rowLo + (size_t)lo1 * IMG_C);
    const float4* p01 = (const float4*)(rowLo + (size_t)hi1 * IMG_C);
    const float4* p10 = (const float4*)(rowHi + (size_t)lo1 * IMG_C);
    const float4* p11 = (const float4*)(rowHi + (size_t)hi1 * IMG_C);

    // Prefetch the hi-row corners (consumed last in the blend chain):
    // lowers to global_prefetch_b8 on gfx1250.
    __builtin_prefetch(p10, 0, 3);
    __builtin_prefetch(p11, 0, 3);

    // Issue all eight b128 gathers back-to-back (single load clause,
    // one s_wait_loadcnt before the FMA blend).
    float4 a00 = p00[0], b00 = p00[1];
    float4 a01 = p01[0], b01 = p01[1];
    float4 a10 = p10[0], b10 = p10[1];
    float4 a11 = p11[0], b11 = p11[1];

    // top = v00*(1-w1) + v01*w1 ; bot = v10*(1-w1) + v11*w1
    // res = top*(1-w0) + bot*w0   (matches reference formulation)
    float4 topA = lerp4(a00, a01, w1, omw1);
    float4 topB = lerp4(b00, b01, w1, omw1);
    float4 botA = lerp4(a10, a11, w1, omw1);
    float4 botB = lerp4(b10, b11, w1, omw1);

    float4 resA = lerp4(topA, botA, w0, omw0);
    float4 resB = lerp4(topB, botB, w0, omw0);

    float4* dst = (float4*)(out + (((size_t)b * GRID0 + y) * GRID1 + x) * IMG_C);
    dst[0] = resA;
    dst[1] = resB;
}

extern "C" void kernel_launch(void* const* d_in, const int* in_sizes, int n_in,
                              void* d_out, int out_size, void* d_ws, size_t ws_size,
                              hipStream_t stream) {
    (void)in_sizes; (void)n_in; (void)out_size; (void)d_ws; (void)ws_size;
    const float* image   = (const float*)d_in[0];   // 64*512*512*8
    const float* section = (const float*)d_in[1];   // 64*3
    float* out = (float*)d_out;                     // 64*64*64*8

    // 64 batches * 16 y-tiles = 1024 blocks; 256 threads = 4 rows x 64 cols
    dim3 grid(1024);
    dim3 block(256);
    bilinear_crop_kernel<<<grid, block, 0, stream>>>(image, section, out);
}